// sRT_global_Optimizer_77730318123476
// MI455X (gfx1250) — compile-verified
//
#include <hip/hip_runtime.h>
#include <math.h>

typedef __attribute__((ext_vector_type(2))) float v2f;
typedef __attribute__((ext_vector_type(8))) float v8f;

// ---------------------------------------------------------------------------
// Kernel A: compose the 8 SE3 pose matrices into one 4x4 transform (tiny).
// One thread does everything; result (row-major 4x4) -> ws[0..15].
// ---------------------------------------------------------------------------
__global__ void compose_kernel(const float* __restrict__ scales,
                               const float* __restrict__ Rm,      // [8,3,3]
                               const float* __restrict__ Tv,      // [8,3]
                               const float* __restrict__ rotd,    // [8,3] (theta)
                               const float* __restrict__ transd,  // [8,3] (rho)
                               float* __restrict__ ws) {
    if (threadIdx.x != 0 || blockIdx.x != 0) return;

    float carry[4][4] = {{1.f,0.f,0.f,0.f},
                         {0.f,1.f,0.f,0.f},
                         {0.f,0.f,1.f,0.f},
                         {0.f,0.f,0.f,1.f}};

    for (int i = 0; i < 8; ++i) {
        const float th0 = rotd[3*i+0], th1 = rotd[3*i+1], th2 = rotd[3*i+2];
        const float rh0 = transd[3*i+0], rh1 = transd[3*i+1], rh2 = transd[3*i+2];

        float W[3][3] = {{0.f, -th2,  th1},
                         {th2,  0.f, -th0},
                         {-th1, th0,  0.f}};
        float W2[3][3];
        #pragma unroll
        for (int r = 0; r < 3; ++r)
            #pragma unroll
            for (int c = 0; c < 3; ++c)
                W2[r][c] = W[r][0]*W[0][c] + W[r][1]*W[1][c] + W[r][2]*W[2][c];

        const float ang = sqrtf(th0*th0 + th1*th1 + th2*th2);
        float Ac, Bc, Dc;
        if (ang < 1e-5f) {
            Ac = 1.f; Bc = 0.5f; Dc = 1.f/6.f;
        } else {
            const float s = sinf(ang), c = cosf(ang);
            Ac = s / ang;
            Bc = (1.f - c) / (ang*ang);
            Dc = (ang - s) / (ang*ang*ang);
        }

        float dR[3][3], V[3][3];
        #pragma unroll
        for (int r = 0; r < 3; ++r)
            #pragma unroll
            for (int c = 0; c < 3; ++c) {
                const float id = (r == c) ? 1.f : 0.f;
                dR[r][c] = id + Ac*W[r][c] + Bc*W2[r][c];   // SO3 exp
                V[r][c]  = id + Bc*W[r][c] + Dc*W2[r][c];   // left Jacobian
            }

        float dt[3];
        #pragma unroll
        for (int r = 0; r < 3; ++r)
            dt[r] = V[r][0]*rh0 + V[r][1]*rh1 + V[r][2]*rh2;

        const float sc = scales[i];
        float M[4][4];
        #pragma unroll
        for (int r = 0; r < 3; ++r) {
            #pragma unroll
            for (int c = 0; c < 3; ++c)
                M[r][c] = sc * (dR[r][0]*Rm[9*i + 0*3 + c] +
                                dR[r][1]*Rm[9*i + 1*3 + c] +
                                dR[r][2]*Rm[9*i + 2*3 + c]);
            M[r][3] = Tv[3*i + r] + dt[r];
        }
        M[3][0] = 0.f; M[3][1] = 0.f; M[3][2] = 0.f; M[3][3] = 1.f;

        // carry = M_i @ carry   (scan => final = M7 @ ... @ M0)
        float nc[4][4];
        #pragma unroll
        for (int r = 0; r < 4; ++r)
            #pragma unroll
            for (int c = 0; c < 4; ++c)
                nc[r][c] = M[r][0]*carry[0][c] + M[r][1]*carry[1][c] +
                           M[r][2]*carry[2][c] + M[r][3]*carry[3][c];
        #pragma unroll
        for (int r = 0; r < 4; ++r)
            #pragma unroll
            for (int c = 0; c < 4; ++c)
                carry[r][c] = nc[r][c];
    }

    #pragma unroll
    for (int r = 0; r < 4; ++r)
        #pragma unroll
        for (int c = 0; c < 4; ++c)
            ws[4*r + c] = carry[r][c];
}

// ---------------------------------------------------------------------------
// Kernel B: out[i] = trans @ x[i] for 8.4M points, via V_WMMA_F32_16X16X4_F32.
// A[16x4] = trans padded (rows 4..15 zero), B[4x16] = 16 point columns,
// D rows 0..3 = outputs, already lane-aligned for a coalesced b128 store.
// Memory-bound: 32 B/point @ 23.3 TB/s.
// ---------------------------------------------------------------------------
__global__ __launch_bounds__(256)
void transform_kernel(const float* __restrict__ xin_f,
                      const float* __restrict__ ws,
                      float* __restrict__ out_f,
                      int npoints) {
    const float4* __restrict__ xin = (const float4*)xin_f;
    float4* __restrict__ o4 = (float4*)out_f;

    // Uniform 4x4 transform (L2-resident after first hit).
    float t[16];
    #pragma unroll
    for (int j = 0; j < 16; ++j) t[j] = ws[j];

    const unsigned lane = threadIdx.x & 31u;   // wave32
    const bool hi = (lane >= 16u);
    const unsigned m = lane & 15u;

    // A operand (16x4 f32, 2 VGPRs): lane m -> A[m][0],A[m][1];
    // lane m+16 -> A[m][2],A[m][3]; rows m>=4 are zero.
    float a0 = 0.f, a1 = 0.f;
    #pragma unroll
    for (int r = 0; r < 4; ++r) {
        const float rk0 = hi ? t[r*4 + 2] : t[r*4 + 0];
        const float rk1 = hi ? t[r*4 + 3] : t[r*4 + 1];
        a0 = (m == (unsigned)r) ? rk0 : a0;
        a1 = (m == (unsigned)r) ? rk1 : a1;
    }
    const v2f Aop = {a0, a1};
    const v8f cz = {0.f,0.f,0.f,0.f,0.f,0.f,0.f,0.f};

    const size_t wave   = (size_t)blockIdx.x * (blockDim.x >> 5) + (threadIdx.x >> 5);
    const size_t nwaves = (size_t)gridDim.x * (blockDim.x >> 5);

    for (size_t base = wave * 32; base < (size_t)npoints; base += nwaves * 32) {
        const float4 p = xin[base + lane];      // lane L holds point base+L

        // Stage B operands: VGPR0 = K0 (lo lanes) / K2 (hi lanes), VGPR1 = K1/K3.
        const float xo = __shfl_xor(p.x, 16, 32);
        const float yo = __shfl_xor(p.y, 16, 32);
        const float zo = __shfl_xor(p.z, 16, 32);
        const float wo = __shfl_xor(p.w, 16, 32);

        const v2f B1 = { hi ? zo  : p.x,  hi ? wo  : p.y };  // points base+0..15
        const v2f B2 = { hi ? p.z : xo,   hi ? p.w : yo  };  // points base+16..31

        v8f d1 = __builtin_amdgcn_wmma_f32_16x16x4_f32(
            false, Aop, false, B1, (short)0, cz, false, false);
        v8f d2 = __builtin_amdgcn_wmma_f32_16x16x4_f32(
            false, Aop, false, B2, (short)0, cz, false, false);

        // d1[r] at lane n (<16) = out[base+n][r]; move d2 results up 16 lanes.
        const float e0 = __shfl_xor(d2[0], 16, 32);
        const float e1 = __shfl_xor(d2[1], 16, 32);
        const float e2 = __shfl_xor(d2[2], 16, 32);
        const float e3 = __shfl_xor(d2[3], 16, 32);

        float4 o;
        o.x = hi ? e0 : d1[0];
        o.y = hi ? e1 : d1[1];
        o.z = hi ? e2 : d1[2];
        o.w = hi ? e3 : d1[3];
        o4[base + lane] = o;                    // fully coalesced b128 store
    }
}

extern "C" void kernel_launch(void* const* d_in, const int* in_sizes, int n_in,
                              void* d_out, int out_size, void* d_ws, size_t ws_size,
                              hipStream_t stream) {
    const float* x       = (const float*)d_in[0];   // [N,4]
    const float* scales  = (const float*)d_in[1];   // [8]
    const float* R       = (const float*)d_in[2];   // [8,3,3]
    const float* T       = (const float*)d_in[3];   // [8,3]
    const float* rotd    = (const float*)d_in[4];   // [8,3]
    const float* transd  = (const float*)d_in[5];   // [8,3]
    float* out = (float*)d_out;
    float* ws  = (float*)d_ws;

    const int npoints = in_sizes[0] / 4;

    compose_kernel<<<1, 32, 0, stream>>>(scales, R, T, rotd, transd, ws);

    // 4096 blocks x 8 waves x 32 pts = 1Mi points/sweep -> 8 uniform iterations.
    transform_kernel<<<4096, 256, 0, stream>>>(x, ws, out, npoints);
}